// MambaBlock_25701084299379
// MI455X (gfx1250) — compile-verified
//
#include <hip/hip_runtime.h>
#include <math.h>

#define NB   2      // batch
#define DM   64     // d_model
#define DI   128    // d_inner
#define NS   16     // d_state
#define RK   4      // dt_rank
#define HH   64
#define WW   64

typedef __attribute__((ext_vector_type(2))) float v2f;
typedef __attribute__((ext_vector_type(8))) float v8f;

__device__ __forceinline__ float siluf(float x) { return x / (1.f + expf(-x)); }
__device__ __forceinline__ float geluf(float x) { return 0.5f * x * (1.f + erff(x * 0.7071067811865476f)); }
__device__ __forceinline__ float softplusf(float x) { return (x > 20.f) ? x : log1pf(expf(x)); }

// ---------------------------------------------------------------------------
// WMMA f32 GEMM (N % 32 == 0 fast path):
//   C[M,N] = alpha * act(A[M,K] * W[N,K]^T + bias[N]) + beta * C
// 128-thread blocks = 4 waves; each wave computes a 16x32 tile (two 16x16
// WMMA accumulators sharing one A fragment per k-step: 3 loads -> 2 wmma).
// K stepped by 4 via v_wmma_f32_16x16x4_f32. M must be a multiple of 16.
// act: 0 = none, 1 = softplus
// ---------------------------------------------------------------------------
__global__ void gemm_wmma2(const float* __restrict__ A, int lda,
                           const float* __restrict__ W, int ldw,
                           const float* __restrict__ bias,
                           float* __restrict__ C, int ldc,
                           int M, int N, int K,
                           float alpha, float beta, int act)
{
    const int tilesN2 = N >> 5;                       // 32-wide column tiles
    const int tile    = blockIdx.x * 4 + (threadIdx.x >> 5);
    if (tile >= (M >> 4) * tilesN2) return;           // wave-uniform
    const int m0   = (tile / tilesN2) << 4;
    const int n0   = (tile % tilesN2) << 5;
    const int lane = threadIdx.x & 31;
    const int half = lane >> 4;      // which K-subpair this lane carries
    const int l16  = lane & 15;

    const int arow  = m0 + l16;      // A: M on lanes 0-15 (mirrored on 16-31)
    const int bcol0 = n0 + l16;      // B: N on lanes
    const int bcol1 = n0 + 16 + l16;

    const float* __restrict__ Ap  = A + (size_t)arow * lda;
    const float* __restrict__ Wp0 = W + (size_t)bcol0 * ldw;
    const float* __restrict__ Wp1 = W + (size_t)bcol1 * ldw;

    v8f acc0 = {0.f, 0.f, 0.f, 0.f, 0.f, 0.f, 0.f, 0.f};
    v8f acc1 = {0.f, 0.f, 0.f, 0.f, 0.f, 0.f, 0.f, 0.f};
    for (int k0 = 0; k0 < K; k0 += 4) {
        const int kk = k0 + half * 2;
        v2f av, bv0, bv1;
        av.x  = Ap[kk];   av.y  = Ap[kk + 1];
        bv0.x = Wp0[kk];  bv0.y = Wp0[kk + 1];
        bv1.x = Wp1[kk];  bv1.y = Wp1[kk + 1];
        acc0 = __builtin_amdgcn_wmma_f32_16x16x4_f32(false, av, false, bv0,
                                                     (short)0, acc0, false, false);
        acc1 = __builtin_amdgcn_wmma_f32_16x16x4_f32(false, av, false, bv1,
                                                     (short)0, acc1, false, false);
    }

    const float bb0 = bias ? bias[bcol0] : 0.f;
    const float bb1 = bias ? bias[bcol1] : 0.f;
#pragma unroll
    for (int v = 0; v < 8; ++v) {
        const int row = m0 + v + half * 8;   // C/D layout: VGPR v -> M = v (+8 hi lanes)
        float v0 = acc0[v] + bb0;
        float v1 = acc1[v] + bb1;
        if (act == 1) { v0 = softplusf(v0); v1 = softplusf(v1); }
        const size_t o0 = (size_t)row * ldc + bcol0;
        const size_t o1 = (size_t)row * ldc + bcol1;
        const float p0 = (beta != 0.f) ? C[o0] : 0.f;
        const float p1 = (beta != 0.f) ? C[o1] : 0.f;
        C[o0] = alpha * v0 + beta * p0;
        C[o1] = alpha * v1 + beta * p1;
    }
}

// ---------------------------------------------------------------------------
// WMMA f32 GEMM, ragged-N path (single 16x16 tile per wave, clamped-address
// + multiplicative zero mask for out-of-range columns; no exec divergence).
// ---------------------------------------------------------------------------
__global__ void gemm_wmma1(const float* __restrict__ A, int lda,
                           const float* __restrict__ W, int ldw,
                           const float* __restrict__ bias,
                           float* __restrict__ C, int ldc,
                           int M, int N, int K,
                           float alpha, float beta, int act)
{
    const int tilesN = (N + 15) >> 4;
    const int tile   = blockIdx.x * 4 + (threadIdx.x >> 5);
    if (tile >= (M >> 4) * tilesN) return;   // wave-uniform
    const int m0   = (tile / tilesN) << 4;
    const int n0   = (tile % tilesN) << 4;
    const int lane = threadIdx.x & 31;
    const int half = lane >> 4;
    const int l16  = lane & 15;

    const int   arow = m0 + l16;
    const int   bcol = n0 + l16;
    const bool  bok  = (bcol < N);
    const float bmsk = bok ? 1.f : 0.f;

    const float* __restrict__ Ap = A + (size_t)arow * lda;
    const float* __restrict__ Wp = W + (size_t)(bok ? bcol : (N - 1)) * ldw; // clamped

    v8f acc = {0.f, 0.f, 0.f, 0.f, 0.f, 0.f, 0.f, 0.f};
    for (int k0 = 0; k0 < K; k0 += 4) {
        const int kk = k0 + half * 2;
        v2f av, bv;
        av.x = Ap[kk];
        av.y = Ap[kk + 1];
        bv.x = Wp[kk] * bmsk;
        bv.y = Wp[kk + 1] * bmsk;
        acc = __builtin_amdgcn_wmma_f32_16x16x4_f32(false, av, false, bv,
                                                    (short)0, acc, false, false);
    }

    if (bok) {
        const float bb = bias ? bias[bcol] : 0.f;
#pragma unroll
        for (int v = 0; v < 8; ++v) {
            const int row = m0 + v + half * 8;
            float val = acc[v] + bb;
            if (act == 1) val = softplusf(val);
            const size_t o = (size_t)row * ldc + bcol;
            const float prev = (beta != 0.f) ? C[o] : 0.f;
            C[o] = alpha * val + beta * prev;
        }
    }
}

// ---------------------------------------------------------------------------
// Causal (or reversed anti-causal) depthwise conv1d width-4 + bias + SiLU.
// Reads the x half (cols [0,DI)) of xz (B,L,2*DI); writes (B,L,DI).
// ---------------------------------------------------------------------------
__global__ void conv1d_silu(const float* __restrict__ xz,
                            const float* __restrict__ w,
                            const float* __restrict__ b,
                            float* __restrict__ out,
                            int B, int L, int rev)
{
    const int idx = blockIdx.x * blockDim.x + threadIdx.x;
    if (idx >= B * L * DI) return;
    const int d  = idx & (DI - 1);
    const int t  = (idx >> 7) % L;
    const int bb = idx / (L * DI);
    const float* xp = xz + (size_t)bb * L * (2 * DI);
    float acc = b[d];
#pragma unroll
    for (int j = 0; j < 4; ++j) {
        const int tt = rev ? (t + 3 - j) : (t - 3 + j);
        if (tt >= 0 && tt < L) acc += w[d * 4 + j] * xp[(size_t)tt * (2 * DI) + d];
    }
    out[idx] = siluf(acc);
}

// ---------------------------------------------------------------------------
// Selective scan. 16 lanes per (b, d_inner) channel; lane = state index n.
// Fuses dA=exp(dt*A), recurrence, C-contraction (shfl_xor reduce over 16
// lanes), + D*x skip, and silu(z) output gate.
// ---------------------------------------------------------------------------
__global__ void sel_scan(const float* __restrict__ dt,
                         const float* __restrict__ xc,
                         const float* __restrict__ proj,
                         const float* __restrict__ xz,
                         const float* __restrict__ A_log,
                         const float* __restrict__ Dp,
                         float* __restrict__ y,
                         int B, int L, int rev)
{
    const int gid = blockIdx.x * blockDim.x + threadIdx.x;
    const int n   = gid & 15;
    const int d   = (gid >> 4) & (DI - 1);
    const int bb  = gid >> 11;
    if (bb >= B) return;   // exact launch: never diverges

    const float An = -expf(A_log[d * NS + n]);
    const float Dd = Dp[d];
    const size_t bx = (size_t)bb * L * DI;
    const size_t bp = (size_t)bb * L * (RK + 2 * NS);
    const size_t bz = (size_t)bb * L * (2 * DI);

    float h = 0.f;
    for (int t = 0; t < L; ++t) {
        const int tt = rev ? (L - 1 - t) : t;
        const float dtv = dt[bx + (size_t)tt * DI + d];
        const float xv  = xc[bx + (size_t)tt * DI + d];
        const float Bv  = proj[bp + (size_t)tt * (RK + 2 * NS) + RK + n];
        const float Cv  = proj[bp + (size_t)tt * (RK + 2 * NS) + RK + NS + n];
        h = h * expf(dtv * An) + dtv * Bv * xv;
        float p = h * Cv;
        p += __shfl_xor(p, 8);
        p += __shfl_xor(p, 4);
        p += __shfl_xor(p, 2);
        p += __shfl_xor(p, 1);
        if (n == 0) {
            const float zv = xz[bz + (size_t)tt * (2 * DI) + DI + d];
            y[bx + (size_t)tt * DI + d] = (p + Dd * xv) * siluf(zv);
        }
    }
}

// ---------------------------------------------------------------------------
// 3x3 same-pad conv, NCHW, full channel mix.
// ---------------------------------------------------------------------------
__global__ void conv3x3(const float* __restrict__ x, const float* __restrict__ w,
                        const float* __restrict__ bias, float* __restrict__ y,
                        int B, int C, int H, int W)
{
    const int idx = blockIdx.x * blockDim.x + threadIdx.x;
    if (idx >= B * C * H * W) return;
    const int wx = idx % W;
    const int hy = (idx / W) % H;
    const int co = (idx / (W * H)) % C;
    const int bb = idx / (W * H * C);
    const float* xb = x + (size_t)bb * C * H * W;
    float acc = bias[co];
    for (int ci = 0; ci < C; ++ci) {
        const float* xc = xb + (size_t)ci * H * W;
        const float* wk = w + ((size_t)co * C + ci) * 9;
#pragma unroll
        for (int kh = 0; kh < 3; ++kh) {
            const int ih = hy + kh - 1;
            if (ih < 0 || ih >= H) continue;
#pragma unroll
            for (int kw = 0; kw < 3; ++kw) {
                const int iw = wx + kw - 1;
                if (iw < 0 || iw >= W) continue;
                acc += wk[kh * 3 + kw] * xc[(size_t)ih * W + iw];
            }
        }
    }
    y[idx] = acc;
}

// ---------------------------------------------------------------------------
// GroupNorm (groups G), optional fused exact GELU. One block per (b, g).
// Safe in-place (read-before-write per element).
// ---------------------------------------------------------------------------
__global__ void groupnorm(const float* __restrict__ x, const float* __restrict__ w,
                          const float* __restrict__ b, float* __restrict__ y,
                          int B, int C, int HW, int G, int do_gelu)
{
    const int g  = blockIdx.x % G;
    const int bb = blockIdx.x / G;
    const int cg = C / G;
    const size_t cnt = (size_t)cg * HW;
    const float* xp = x + ((size_t)bb * C + (size_t)g * cg) * HW;
    float* yp = y + ((size_t)bb * C + (size_t)g * cg) * HW;

    float s = 0.f, ss = 0.f;
    for (size_t i = threadIdx.x; i < cnt; i += blockDim.x) {
        const float v = xp[i];
        s += v; ss += v * v;
    }
    __shared__ float sh[512];
    sh[threadIdx.x] = s;
    sh[256 + threadIdx.x] = ss;
    __syncthreads();
    for (int st = 128; st > 0; st >>= 1) {
        if ((int)threadIdx.x < st) {
            sh[threadIdx.x] += sh[threadIdx.x + st];
            sh[256 + threadIdx.x] += sh[256 + threadIdx.x + st];
        }
        __syncthreads();
    }
    const float m   = sh[0] / (float)cnt;
    const float var = sh[256] / (float)cnt - m * m;
    const float inv = rsqrtf(var + 1e-5f);
    for (size_t i = threadIdx.x; i < cnt; i += blockDim.x) {
        const int c = g * cg + (int)(i / HW);
        float v = (xp[i] - m) * inv * w[c] + b[c];
        if (do_gelu) v = geluf(v);
        yp[i] = v;
    }
}

__global__ void dwconv3x3_gelu(const float* __restrict__ x, const float* __restrict__ w,
                               const float* __restrict__ b, float* __restrict__ y,
                               int B, int C, int H, int W)
{
    const int idx = blockIdx.x * blockDim.x + threadIdx.x;
    if (idx >= B * C * H * W) return;
    const int wx = idx % W;
    const int hy = (idx / W) % H;
    const int c  = (idx / (W * H)) % C;
    const int bb = idx / (W * H * C);
    const float* xc = x + ((size_t)bb * C + c) * H * W;
    const float* wk = w + (size_t)c * 9;
    float acc = b[c];
#pragma unroll
    for (int kh = 0; kh < 3; ++kh) {
        const int ih = hy + kh - 1;
        if (ih < 0 || ih >= H) continue;
#pragma unroll
        for (int kw = 0; kw < 3; ++kw) {
            const int iw = wx + kw - 1;
            if (iw < 0 || iw >= W) continue;
            acc += wk[kh * 3 + kw] * xc[(size_t)ih * W + iw];
        }
    }
    y[idx] = geluf(acc);
}

__global__ void pwconv(const float* __restrict__ x, const float* __restrict__ w,
                       const float* __restrict__ b, float* __restrict__ y,
                       int B, int C, int HW)
{
    const int idx = blockIdx.x * blockDim.x + threadIdx.x;
    if (idx >= B * C * HW) return;
    const int p  = idx % HW;
    const int co = (idx / HW) % C;
    const int bb = idx / (HW * C);
    const float* xb = x + (size_t)bb * C * HW + p;
    float acc = b[co];
    for (int ci = 0; ci < C; ++ci) acc += w[co * C + ci] * xb[(size_t)ci * HW];
    y[idx] = acc;
}

__global__ void add_gelu(const float* __restrict__ a, const float* __restrict__ b,
                         float* __restrict__ y, int n)
{
    const int idx = blockIdx.x * blockDim.x + threadIdx.x;
    if (idx < n) y[idx] = geluf(a[idx] + b[idx]);
}

__global__ void nchw2nlc(const float* __restrict__ in, float* __restrict__ out,
                         int B, int C, int HW)
{
    const int idx = blockIdx.x * blockDim.x + threadIdx.x;
    if (idx >= B * HW * C) return;
    const int c  = idx % C;
    const int l  = (idx / C) % HW;
    const int bb = idx / (C * HW);
    out[idx] = in[((size_t)bb * C + c) * HW + l];
}

__global__ void nlc2nchw(const float* __restrict__ in, float* __restrict__ out,
                         int B, int C, int HW)
{
    const int idx = blockIdx.x * blockDim.x + threadIdx.x;
    if (idx >= B * C * HW) return;
    const int l  = idx % HW;
    const int c  = (idx / HW) % C;
    const int bb = idx / (C * HW);
    out[idx] = in[((size_t)bb * HW + l) * C + c];
}

__global__ void avgpool2(const float* __restrict__ x, float* __restrict__ y,
                         int B, int C, int H, int W)
{
    const int Ho = H / 2, Wo = W / 2;
    const int idx = blockIdx.x * blockDim.x + threadIdx.x;
    if (idx >= B * C * Ho * Wo) return;
    const int wo = idx % Wo;
    const int ho = (idx / Wo) % Ho;
    const int c  = (idx / (Wo * Ho)) % C;
    const int bb = idx / (Wo * Ho * C);
    const float* xc = x + ((size_t)bb * C + c) * H * W;
    const int h2 = ho * 2, w2 = wo * 2;
    y[idx] = 0.25f * (xc[(size_t)h2 * W + w2]     + xc[(size_t)h2 * W + w2 + 1] +
                      xc[(size_t)(h2+1) * W + w2] + xc[(size_t)(h2+1) * W + w2 + 1]);
}

// half-pixel-center bilinear 2x upsample (matches jax.image.resize 'bilinear')
__global__ void upsample2x(const float* __restrict__ in, float* __restrict__ out,
                           int B, int C, int Hi, int Wi)
{
    const int Ho = Hi * 2, Wo = Wi * 2;
    const int idx = blockIdx.x * blockDim.x + threadIdx.x;
    if (idx >= B * C * Ho * Wo) return;
    const int wo = idx % Wo;
    const int ho = (idx / Wo) % Ho;
    const int c  = (idx / (Wo * Ho)) % C;
    const int bb = idx / (Wo * Ho * C);
    const float sh = ho * 0.5f - 0.25f;
    const float sw = wo * 0.5f - 0.25f;
    int h0 = (int)floorf(sh); const float fh = sh - (float)h0;
    int w0 = (int)floorf(sw); const float fw = sw - (float)w0;
    const int h0c = min(max(h0, 0), Hi - 1);
    const int h1c = min(max(h0 + 1, 0), Hi - 1);
    const int w0c = min(max(w0, 0), Wi - 1);
    const int w1c = min(max(w0 + 1, 0), Wi - 1);
    const float* xc = in + ((size_t)bb * C + c) * Hi * Wi;
    out[idx] = (1.f - fh) * ((1.f - fw) * xc[(size_t)h0c * Wi + w0c] + fw * xc[(size_t)h0c * Wi + w1c])
             +        fh  * ((1.f - fw) * xc[(size_t)h1c * Wi + w0c] + fw * xc[(size_t)h1c * Wi + w1c]);
}

__global__ void combine4(const float* __restrict__ xs, const float* __restrict__ xl,
                         const float* __restrict__ xtg, const float* __restrict__ xtd,
                         float* __restrict__ out, int n)
{
    const int idx = blockIdx.x * blockDim.x + threadIdx.x;
    if (idx < n) out[idx] = xs[idx] + xl[idx] + 0.5f * (xtg[idx] + xtd[idx]);
}

// ---------------------------------------------------------------------------
// Host side
// ---------------------------------------------------------------------------
struct MParams {
    const float *in_w, *cw, *cb, *xp, *dtw, *dtb, *al, *dp, *ow;
};

static inline void gemm_launch(hipStream_t s, const float* A, int lda,
                               const float* W, int ldw, const float* bias,
                               float* C, int ldc, int M, int N, int K,
                               float alpha, float beta, int act)
{
    if ((N & 31) == 0) {
        const int tiles  = (M / 16) * (N / 32);
        const int blocks = (tiles + 3) / 4;          // 4 waves (16x32 tiles) per block
        gemm_wmma2<<<dim3(blocks), dim3(128), 0, s>>>(A, lda, W, ldw, bias, C, ldc,
                                                      M, N, K, alpha, beta, act);
    } else {
        const int tiles  = (M / 16) * ((N + 15) / 16);
        const int blocks = (tiles + 3) / 4;          // 4 waves (16x16 tiles) per block
        gemm_wmma1<<<dim3(blocks), dim3(128), 0, s>>>(A, lda, W, ldw, bias, C, ldc,
                                                      M, N, K, alpha, beta, act);
    }
}

// One mamba direction: in_proj -> conv1d+silu -> x_proj -> dt(softplus) ->
// scan (+gate) -> out_proj (alpha=0.5, beta accumulates the two directions).
static void mamba_dir(hipStream_t s, const float* u, int B, int L, const MParams& p,
                      int rev, float beta, float* acc,
                      float* xz, float* xconv, float* proj, float* dtb, float* ysc)
{
    const int M = B * L;
    gemm_launch(s, u, DM, p.in_w, DM, nullptr, xz, 2 * DI, M, 2 * DI, DM, 1.f, 0.f, 0);
    const int tot = M * DI;
    conv1d_silu<<<(tot + 255) / 256, 256, 0, s>>>(xz, p.cw, p.cb, xconv, B, L, rev);
    gemm_launch(s, xconv, DI, p.xp, DI, nullptr, proj, RK + 2 * NS, M, RK + 2 * NS, DI, 1.f, 0.f, 0);
    gemm_launch(s, proj, RK + 2 * NS, p.dtw, RK, p.dtb, dtb, DI, M, DI, RK, 1.f, 0.f, 1);
    sel_scan<<<(B * DI * NS + 255) / 256, 256, 0, s>>>(dtb, xconv, proj, xz, p.al, p.dp, ysc, B, L, rev);
    gemm_launch(s, ysc, DI, p.ow, DI, nullptr, acc, DM, M, DM, DI, 0.5f, beta, 0);
}

extern "C" void kernel_launch(void* const* d_in, const int* in_sizes, int n_in,
                              void* d_out, int out_size, void* d_ws, size_t ws_size,
                              hipStream_t stream)
{
    (void)in_sizes; (void)n_in; (void)out_size; (void)ws_size;

    const float* x        = (const float*)d_in[0];
    const float* m_in_w   = (const float*)d_in[1];
    const float* m_conv_w = (const float*)d_in[2];
    const float* m_conv_b = (const float*)d_in[3];
    const float* m_xproj  = (const float*)d_in[4];
    const float* m_dt_w   = (const float*)d_in[5];
    const float* m_dt_b   = (const float*)d_in[6];
    const float* m_A_log  = (const float*)d_in[7];
    const float* m_D      = (const float*)d_in[8];
    const float* m_out_w  = (const float*)d_in[9];
    const float* dw_w = (const float*)d_in[10];
    const float* dw_b = (const float*)d_in[11];
    const float* pw_w = (const float*)d_in[12];
    const float* pw_b = (const float*)d_in[13];
    const float* r1c1w = (const float*)d_in[14];
    const float* r1c1b = (const float*)d_in[15];
    const float* r1g1w = (const float*)d_in[16];
    const float* r1g1b = (const float*)d_in[17];
    const float* r1c2w = (const float*)d_in[18];
    const float* r1c2b = (const float*)d_in[19];
    const float* r1g2w = (const float*)d_in[20];
    const float* r1g2b = (const float*)d_in[21];
    const float* r2c1w = (const float*)d_in[22];
    const float* r2c1b = (const float*)d_in[23];
    const float* r2g1w = (const float*)d_in[24];
    const float* r2g1b = (const float*)d_in[25];
    const float* r2c2w = (const float*)d_in[26];
    const float* r2c2b = (const float*)d_in[27];
    const float* r2g2w = (const float*)d_in[28];
    const float* r2g2b = (const float*)d_in[29];

    const int HW = HH * WW;             // 4096
    const int L  = HW;
    const int L2 = (HH / 2) * (WW / 2); // 1024
    const size_t SZ = (size_t)NB * DM * HW; // 524288

    float* ws = (float*)d_ws;
    float* xflat  = ws;
    float* acc    = ws + 1 * SZ;
    float* xs     = ws + 2 * SZ;   // spatial mamba, NCHW
    float* xtg    = ws + 3 * SZ;   // time mamba, NCHW
    float* xtd    = ws + 4 * SZ;   // down-up mamba, NCHW
    float* t0     = ws + 5 * SZ;
    float* t1     = ws + 6 * SZ;
    float* xl     = ws + 7 * SZ;
    float* xz     = ws + 8 * SZ;                      // B*L*256
    float* xconv  = xz + (size_t)NB * L * 2 * DI;     // B*L*128
    float* proj   = xconv + (size_t)NB * L * DI;      // B*L*36
    float* dtb    = proj + (size_t)NB * L * (RK + 2 * NS);
    float* ysc    = dtb + (size_t)NB * L * DI;
    float* dsflat = ysc + (size_t)NB * L * DI;        // B*1024*64
    float* tdsm   = dsflat + (size_t)NB * L2 * DM;    // B*64*32*32

    MParams mp[6];
    for (int i = 0; i < 6; ++i) {
        mp[i].in_w = m_in_w   + (size_t)i * 2 * DI * DM;
        mp[i].cw   = m_conv_w + (size_t)i * DI * 4;
        mp[i].cb   = m_conv_b + (size_t)i * DI;
        mp[i].xp   = m_xproj  + (size_t)i * (RK + 2 * NS) * DI;
        mp[i].dtw  = m_dt_w   + (size_t)i * DI * RK;
        mp[i].dtb  = m_dt_b   + (size_t)i * DI;
        mp[i].al   = m_A_log  + (size_t)i * DI * NS;
        mp[i].dp   = m_D      + (size_t)i * DI;
        mp[i].ow   = m_out_w  + (size_t)i * DM * DI;
    }

    const int nelem = (int)SZ;
    const dim3 gE((nelem + 255) / 256), bE(256);

    // x (NCHW) -> x_flat (B, L, D)
    nchw2nlc<<<gE, bE, 0, stream>>>(x, xflat, NB, DM, HW);

    // ---- spatial bidirectional mamba -> xs (NCHW) ----
    mamba_dir(stream, xflat, NB, L, mp[0], 0, 0.f, acc, xz, xconv, proj, dtb, ysc);
    mamba_dir(stream, xflat, NB, L, mp[1], 1, 1.f, acc, xz, xconv, proj, dtb, ysc);
    nlc2nchw<<<gE, bE, 0, stream>>>(acc, xs, NB, DM, HW);

    // ---- local conv branch on xs -> xl ----
    conv3x3<<<gE, bE, 0, stream>>>(xs, r1c1w, r1c1b, t0, NB, DM, HH, WW);
    groupnorm<<<dim3(NB * 4), dim3(256), 0, stream>>>(t0, r1g1w, r1g1b, t0, NB, DM, HW, 4, 1);
    conv3x3<<<gE, bE, 0, stream>>>(t0, r1c2w, r1c2b, t1, NB, DM, HH, WW);
    groupnorm<<<dim3(NB * 4), dim3(256), 0, stream>>>(t1, r1g2w, r1g2b, t1, NB, DM, HW, 4, 0);
    add_gelu<<<gE, bE, 0, stream>>>(xs, t1, xl, nelem);
    dwconv3x3_gelu<<<gE, bE, 0, stream>>>(xl, dw_w, dw_b, t0, NB, DM, HH, WW);
    pwconv<<<gE, bE, 0, stream>>>(t0, pw_w, pw_b, t1, NB, DM, HW);
    conv3x3<<<gE, bE, 0, stream>>>(t1, r2c1w, r2c1b, t0, NB, DM, HH, WW);
    groupnorm<<<dim3(NB * 4), dim3(256), 0, stream>>>(t0, r2g1w, r2g1b, t0, NB, DM, HW, 4, 1);
    conv3x3<<<gE, bE, 0, stream>>>(t0, r2c2w, r2c2b, xl, NB, DM, HH, WW);   // xl no longer needed as input
    groupnorm<<<dim3(NB * 4), dim3(256), 0, stream>>>(xl, r2g2w, r2g2b, xl, NB, DM, HW, 4, 0);
    add_gelu<<<gE, bE, 0, stream>>>(t1, xl, xl, nelem);                     // final conv-branch output

    // ---- 'time' bidirectional mamba on original x_flat -> xtg (NCHW) ----
    mamba_dir(stream, xflat, NB, L, mp[2], 0, 0.f, acc, xz, xconv, proj, dtb, ysc);
    mamba_dir(stream, xflat, NB, L, mp[3], 1, 1.f, acc, xz, xconv, proj, dtb, ysc);
    nlc2nchw<<<gE, bE, 0, stream>>>(acc, xtg, NB, DM, HW);

    // ---- downsampled bidirectional mamba -> upsample -> xtd (NCHW) ----
    const int nds = NB * DM * L2;
    avgpool2<<<(nds + 255) / 256, 256, 0, stream>>>(x, t0, NB, DM, HH, WW);
    nchw2nlc<<<(nds + 255) / 256, 256, 0, stream>>>(t0, dsflat, NB, DM, L2);
    mamba_dir(stream, dsflat, NB, L2, mp[4], 0, 0.f, acc, xz, xconv, proj, dtb, ysc);
    mamba_dir(stream, dsflat, NB, L2, mp[5], 1, 1.f, acc, xz, xconv, proj, dtb, ysc);
    nlc2nchw<<<(nds + 255) / 256, 256, 0, stream>>>(acc, tdsm, NB, DM, L2);
    upsample2x<<<gE, bE, 0, stream>>>(tdsm, xtd, NB, DM, HH / 2, WW / 2);

    // ---- final combine ----
    combine4<<<gE, bE, 0, stream>>>(xs, xl, xtg, xtd, (float*)d_out, nelem);
}